// LatentAugmenter_83777632076050
// MI455X (gfx1250) — compile-verified
//
#include <hip/hip_runtime.h>
#include <hip/hip_bf16.h>

// ---------------------------------------------------------------------------
// LatentAugmenter fused kernel for MI455X (gfx1250, wave32).
// One workgroup (256 threads = 8 waves) per (b, c) 64x64 plane.
//   1) async-stage input plane -> LDS (global_load_async_to_lds_b128)
//   2) fused crop-resize / circular-shift / patch-shuffle gather -> y in LDS
//   3) high-pass via two 64x64x64 f32 GEMMs (Z = A*Y*A^T) on v_wmma_f32_16x16x4_f32,
//      subtracted in place into y
//   4) async-stream y -> out (global_store_async_from_lds_b128)
// All per-sample conditions are workgroup-uniform -> EXEC all ones for WMMA.
// ---------------------------------------------------------------------------

typedef float v8f __attribute__((ext_vector_type(8)));
typedef float v2f __attribute__((ext_vector_type(2)));

#define BB 512
#define CC 16
#define HH 64
#define WW 64
#define NPIX (HH * WW)

#if __has_builtin(__builtin_amdgcn_global_load_async_to_lds_b128) && \
    __has_builtin(__builtin_amdgcn_global_store_async_from_lds_b128)
#define HAVE_ASYNC_LDS 1
#else
#define HAVE_ASYNC_LDS 0
#endif

#if HAVE_ASYNC_LDS
// Parameter 0 is diagnosed by hipcc as 'int __attribute__((vector_size(16))) *'
// (generic AS). Use an int4 vector pointer for the global side and an
// addrspace(3)-qualified int4 pointer for the LDS side (implicitly convertible
// to any of the plausible declared LDS parameter types).
typedef int v4i_b128 __attribute__((vector_size(16)));
typedef __attribute__((address_space(3))) v4i_b128 lds_v4i;
#endif

__device__ __forceinline__ void wait_async0() {
#if HAVE_ASYNC_LDS
#if __has_builtin(__builtin_amdgcn_s_wait_asynccnt)
    __builtin_amdgcn_s_wait_asynccnt(0);
#else
    asm volatile("s_wait_asynccnt 0x0" ::: "memory");
#endif
#endif
}

__device__ __forceinline__ int imin(int a, int b) { return a < b ? a : b; }

// C(16x16) = sum_k L[16*ti + m][k] * R[k][16*tj + n]  (k = 0..63)
// R_T = true means R is stored transposed (we index R[n_row][k]).
template <bool R_T>
__device__ __forceinline__ v8f gemm64_tile(const float* __restrict__ L,
                                           const float* __restrict__ R,
                                           int ti, int tj, int lane) {
    v8f acc = {};
    const int m = lane & 15;
    const int n = lane & 15;
    const int khalf = (lane >> 4) << 1;  // 0 or 2
    const float* lrow = L + (ti * 16 + m) * 64;
    const float* rrow = R_T ? (R + (tj * 16 + n) * 64) : nullptr;
#pragma unroll
    for (int k0 = 0; k0 < 64; k0 += 4) {
        const int ka = k0 + khalf;
        v2f a, b;
        a.x = lrow[ka];
        a.y = lrow[ka + 1];
        if (R_T) {
            b.x = rrow[ka];
            b.y = rrow[ka + 1];
        } else {
            b.x = R[ka * 64 + tj * 16 + n];
            b.y = R[(ka + 1) * 64 + tj * 16 + n];
        }
        acc = __builtin_amdgcn_wmma_f32_16x16x4_f32(
            /*neg_a=*/false, a, /*neg_b=*/false, b,
            /*c_mod=*/(short)0, acc, /*reuse_a=*/false, /*reuse_b=*/false);
    }
    return acc;
}

__global__ __launch_bounds__(256) void latent_aug_kernel(
    const float* __restrict__ x,
    const float* __restrict__ u_crop,
    const float* __restrict__ u_scale,
    const float* __restrict__ u_top,
    const float* __restrict__ u_left,
    const float* __restrict__ u_perm,
    const float* __restrict__ u_hp,
    const int* __restrict__ dy_raw,
    const int* __restrict__ dx_raw,
    float* __restrict__ out) {
    __shared__ float xin[NPIX];   // input plane; reused as GEMM1 temp T
    __shared__ float ybuf[NPIX];  // post step-3 plane; finalized in place
    __shared__ float Amat[NPIX];  // 64x64 lowpass operator
    __shared__ int ci0[HH];       // crop bilinear tables (axis-shared: nh == nw)
    __shared__ int ci1[HH];
    __shared__ float cw[HH];
    __shared__ int ui0[HH];       // lowpass bilinear-up tables
    __shared__ int ui1[HH];
    __shared__ float uw[HH];

    const int tid = threadIdx.x;
    const int plane = blockIdx.x;  // b*16 + c, matches x layout
    const int b = plane >> 4;
    const float* __restrict__ gsrc = x + (size_t)plane * NPIX;
    float* __restrict__ outp = out + (size_t)plane * NPIX;

    // ---- stage input plane into LDS (16 KB contiguous) ----
#if HAVE_ASYNC_LDS
    for (int i = tid; i < NPIX / 4; i += 256) {
        __builtin_amdgcn_global_load_async_to_lds_b128(
            (v4i_b128*)(gsrc + i * 4), (lds_v4i*)(xin + i * 4), 0, 0);
    }
#else
    {
        const float4* s4 = (const float4*)gsrc;
        float4* d4 = (float4*)xin;
        for (int i = tid; i < NPIX / 4; i += 256) d4[i] = s4[i];
    }
#endif

    // ---- per-sample uniform scalars (overlapped with async staging) ----
    const float uc = u_crop[b];
    const float us = u_scale[b];
    const float ut = u_top[b];
    const float ul = u_left[b];
    const float uh = u_hp[b];
    const int dy = dy_raw[b] - 2;
    const int dx = dx_raw[b] - 2;

    const float scale = 0.7f + us * 0.3f;
    int nh = (int)floorf(64.0f * scale);
    if (nh < 2) nh = 2;
    const int top = (int)floorf(ut * (float)(64 - nh + 1));
    const int left = (int)floorf(ul * (float)(64 - nh + 1));

    const bool do_crop = uc < 0.5f;
    float tsh = ut + ul;
    tsh -= floorf(tsh);
    const bool do_shift = tsh < 0.5f;
    float tps = uc + us;
    tps -= floorf(tps);
    const bool do_ps = tps < 0.3f;
    const bool do_hp = uh < 0.3f;

    // stable argsort of 4 floats -> perm (index of k-th smallest)
    float up4[4];
#pragma unroll
    for (int j = 0; j < 4; ++j) up4[j] = u_perm[b * 4 + j];
    int perm[4];
#pragma unroll
    for (int k = 0; k < 4; ++k) {
        int r = 0;
#pragma unroll
        for (int j = 0; j < 4; ++j)
            r += (up4[j] < up4[k]) || (up4[j] == up4[k] && j < k);
        perm[r] = k;
    }

    // ---- interpolation tables (threads 0..63) ----
    if (tid < HH) {
        const float nf = (float)nh;
        float s = ((float)tid + 0.5f) * nf * (1.0f / 64.0f) - 0.5f;
        s = fminf(fmaxf(s, 0.0f), nf - 1.0f);
        const int i0 = (int)floorf(s);
        ci0[tid] = i0;
        ci1[tid] = imin(i0 + 1, nh - 1);
        cw[tid] = s - (float)i0;

        float s2 = ((float)tid + 0.5f) * 0.125f - 0.5f;
        s2 = fminf(fmaxf(s2, 0.0f), 7.0f);
        const int j0 = (int)floorf(s2);
        ui0[tid] = j0;
        ui1[tid] = imin(j0 + 1, 7);
        uw[tid] = s2 - (float)j0;
    }

    wait_async0();   // per-wave: async LDS writes landed before publishing
    __syncthreads();

    // ---- steps 1-3 fused: gather from xin -> ybuf ----
    for (int idx = tid; idx < NPIX; idx += 256) {
        int h = idx >> 6;
        int w = idx & 63;
        if (do_ps) {
            const int q = (((h >> 5) << 1) | (w >> 5));
            const int qs = perm[q];
            h = ((qs >> 1) << 5) | (h & 31);
            w = ((qs & 1) << 5) | (w & 31);
        }
        if (do_shift) {
            h = (h - dy + 64) & 63;
            w = (w - dx + 64) & 63;
        }
        float v;
        if (do_crop) {
            const int r0 = top + ci0[h], r1 = top + ci1[h];
            const float fr = cw[h];
            const int c0 = left + ci0[w], c1 = left + ci1[w];
            const float fc = cw[w];
            const float a00 = xin[(r0 << 6) | c0];
            const float a01 = xin[(r0 << 6) | c1];
            const float a10 = xin[(r1 << 6) | c0];
            const float a11 = xin[(r1 << 6) | c1];
            const float rtop = a00 * (1.0f - fc) + a01 * fc;
            const float rbot = a10 * (1.0f - fc) + a11 * fc;
            v = rtop * (1.0f - fr) + rbot * fr;
        } else {
            v = xin[(h << 6) | w];
        }
        ybuf[idx] = v;
    }

    // ---- build lowpass operator A (only on hp path; workgroup-uniform) ----
    if (do_hp) {
        for (int idx = tid; idx < NPIX; idx += 256) {
            const int o = idx >> 6;
            const int kb = (idx & 63) >> 3;
            float wv = 0.0f;
            if (kb == ui0[o]) wv += 1.0f - uw[o];
            if (kb == ui1[o]) wv += uw[o];
            Amat[idx] = 0.125f * wv;
        }
    }
    __syncthreads();

    const int wave = tid >> 5;
    const int lane = tid & 31;

    if (do_hp) {
        // GEMM1: T = A * Y  -> xin  (xin's input role is done)
        for (int t = wave; t < 16; t += 8) {
            const int ti = t >> 2, tj = t & 3;
            v8f acc = gemm64_tile<false>(Amat, ybuf, ti, tj, lane);
#pragma unroll
            for (int v = 0; v < 8; ++v) {
                const int mg = ti * 16 + v + ((lane >> 4) << 3);
                const int ng = tj * 16 + (lane & 15);
                xin[mg * 64 + ng] = acc[v];
            }
        }
        __syncthreads();
        // GEMM2: Z = T * A^T ; subtract in place into ybuf (disjoint ownership)
        for (int t = wave; t < 16; t += 8) {
            const int ti = t >> 2, tj = t & 3;
            v8f acc = gemm64_tile<true>(xin, Amat, ti, tj, lane);
#pragma unroll
            for (int v = 0; v < 8; ++v) {
                const int mg = ti * 16 + v + ((lane >> 4) << 3);
                const int ng = tj * 16 + (lane & 15);
                const int idx = mg * 64 + ng;
                ybuf[idx] = ybuf[idx] - acc[v];
            }
        }
    }
    __syncthreads();

    // ---- stream finalized plane LDS -> HBM, fully coalesced b128 ----
#if HAVE_ASYNC_LDS
    for (int i = tid; i < NPIX / 4; i += 256) {
        __builtin_amdgcn_global_store_async_from_lds_b128(
            (v4i_b128*)(outp + i * 4), (lds_v4i*)(ybuf + i * 4), 0, 0);
    }
    wait_async0();
#else
    {
        float4* o4 = (float4*)outp;
        const float4* y4 = (const float4*)ybuf;
        for (int i = tid; i < NPIX / 4; i += 256) o4[i] = y4[i];
    }
#endif
}

extern "C" void kernel_launch(void* const* d_in, const int* in_sizes, int n_in,
                              void* d_out, int out_size, void* d_ws, size_t ws_size,
                              hipStream_t stream) {
    (void)in_sizes; (void)n_in; (void)out_size; (void)d_ws; (void)ws_size;
    const float* x       = (const float*)d_in[0];
    const float* u_crop  = (const float*)d_in[1];
    const float* u_scale = (const float*)d_in[2];
    const float* u_top   = (const float*)d_in[3];
    const float* u_left  = (const float*)d_in[4];
    const float* u_perm  = (const float*)d_in[5];
    const float* u_hp    = (const float*)d_in[6];
    const int*   dy_raw  = (const int*)d_in[7];
    const int*   dx_raw  = (const int*)d_in[8];
    float* out = (float*)d_out;

    const int nblocks = BB * CC;  // one (b, c) plane per workgroup
    latent_aug_kernel<<<nblocks, 256, 0, stream>>>(
        x, u_crop, u_scale, u_top, u_left, u_perm, u_hp, dy_raw, dx_raw, out);
}